// SPELL_bak0827_53953379173218
// MI455X (gfx1250) — compile-verified
//
#include <hip/hip_runtime.h>
#include <hip/hip_bf16.h>

// ---------------------------------------------------------------------------
// SPELL GNN forward, collapsed-linear formulation for MI455X (gfx1250).
//
// The whole pre-segment_max pipeline is LINEAR, so it collapses to 8 scalars
// per node:  v[node] = x[node,0:256] @ W[256,8] + spk/spa/bias terms.
// Node GEMM runs on fp32 WMMA (v_wmma_f32_16x16x4_f32) -> exact f32 math.
//   - B operand (collapsed weights) staged in LDS as float2 pairs: one
//     ds_load_b64 per chunk instead of two strided global_load_b32.
//   - A operand software-pipelined: chunk k+4 prefetched while WMMA on k runs.
// Edge phase = 16B gather + encoded-u32 atomicMax segment reduction (L2-res).
// Total traffic ~60MB -> ~3us at 23.3 TB/s; fully bandwidth-bound by design.
// ---------------------------------------------------------------------------

typedef __attribute__((ext_vector_type(2))) float v2f;
typedef __attribute__((ext_vector_type(8))) float v8f;

#define N_NODES 50000
#define N_EDGES 800000
#define XDIM    901
#define TILES   (N_NODES / 16)   // 3125, exact

// Order-preserving IEEE-754 -> u32 encoding (monotone bijection).
// encf(-inf)=0x007FFFFF > 0, so 0 is a safe "empty segment" sentinel.
__device__ __forceinline__ unsigned int encf(float f) {
  unsigned int u = __float_as_uint(f);
  return (u & 0x80000000u) ? ~u : (u | 0x80000000u);
}
__device__ __forceinline__ float decf(unsigned int u) {
  unsigned int b = (u & 0x80000000u) ? (u ^ 0x80000000u) : ~u;
  return __uint_as_float(b);
}

// ---------------------------------------------------------------------------
// K0: fold all weight matrices into the collapsed per-node operator.
//   Global output columns g: 0,1 = aFace ; 2,3 = bFace ; 4,5 = aBody ; 6,7 = bBody
//   tu(j,g): final 64->1 projection; a-cols use (ew_top - ew_bot), b-cols ew_bot.
// ---------------------------------------------------------------------------
__global__ __launch_bounds__(256)
void spell_collapse(const float* __restrict__ sw,  const float* __restrict__ sb,
                    const float* __restrict__ pw,  const float* __restrict__ pb,
                    const float* __restrict__ f1w, const float* __restrict__ f1b,
                    const float* __restrict__ f2w, const float* __restrict__ f2b,
                    const float* __restrict__ few, const float* __restrict__ feb,
                    const float* __restrict__ g1w, const float* __restrict__ g1b,
                    const float* __restrict__ g2w, const float* __restrict__ g2b,
                    const float* __restrict__ gew, const float* __restrict__ geb,
                    float* __restrict__ Wpad, float* __restrict__ Sc,
                    float* __restrict__ Pc,   float* __restrict__ Cb)
{
  const int tid = threadIdx.x;
  for (int i = tid; i < 256 * 16; i += 256) Wpad[i] = 0.0f;
  __syncthreads();

  // Main collapsed weights: 128 x 8 entries, each a 64-dot.
  for (int e = tid; e < 1024; e += 256) {
    const int g = e & 7, k = e >> 3;
    const int blk = g >> 2, c = g & 1;
    const bool isA = (g & 2) == 0;
    const float* ew = blk ? gew : few;
    const float* w1 = blk ? g1w : f1w;
    const float* w2 = blk ? g2w : f2w;
    float s = 0.0f;
    for (int j = 0; j < 64; ++j) {
      float tu = ew[(64 + j) * 2 + c];
      if (isA) tu = ew[j * 2 + c] - tu;
      s += (w1[k * 64 + j] + w2[k * 64 + j]) * tu;
    }
    Wpad[(blk ? 128 + k : k) * 16 + g] = s;
  }

  // Speaker table Sc[3][8]: speaker_w @ f2_w[128:144] @ tu
  for (int e = tid; e < 24; e += 256) {
    const int g = e & 7, spk = e >> 3;
    const int blk = g >> 2, c = g & 1;
    const bool isA = (g & 2) == 0;
    const float* ew = blk ? gew : few;
    const float* w2 = blk ? g2w : f2w;
    float acc = 0.0f;
    for (int s16 = 0; s16 < 16; ++s16) {
      float q = 0.0f;
      for (int j = 0; j < 64; ++j) {
        float tu = ew[(64 + j) * 2 + c];
        if (isA) tu = ew[j * 2 + c] - tu;
        q += w2[(128 + s16) * 64 + j] * tu;
      }
      acc += sw[spk * 16 + s16] * q;
    }
    Sc[spk * 8 + g] = acc;
  }

  // Spatial table Pc[4][8]: spatial_w @ f2_w[144:160] @ tu
  for (int e = tid; e < 32; e += 256) {
    const int g = e & 7, p = e >> 3;
    const int blk = g >> 2, c = g & 1;
    const bool isA = (g & 2) == 0;
    const float* ew = blk ? gew : few;
    const float* w2 = blk ? g2w : f2w;
    float acc = 0.0f;
    for (int s16 = 0; s16 < 16; ++s16) {
      float q = 0.0f;
      for (int j = 0; j < 64; ++j) {
        float tu = ew[(64 + j) * 2 + c];
        if (isA) tu = ew[j * 2 + c] - tu;
        q += w2[(144 + s16) * 64 + j] * tu;
      }
      acc += pw[p * 16 + s16] * q;
    }
    Pc[p * 8 + g] = acc;
  }

  // Bias Cb[8]: (f1_b + f2_b + speaker_b@.. + spatial_b@..) @ tu (+ edge bias on a-cols)
  for (int e = tid; e < 8; e += 256) {
    const int g = e;
    const int blk = g >> 2, c = g & 1;
    const bool isA = (g & 2) == 0;
    const float* ew  = blk ? gew : few;
    const float* b1  = blk ? g1b : f1b;
    const float* b2  = blk ? g2b : f2b;
    const float* w2  = blk ? g2w : f2w;
    const float* eb  = blk ? geb : feb;
    float s = 0.0f;
    for (int j = 0; j < 64; ++j) {
      float hb = b1[j] + b2[j];
      for (int s16 = 0; s16 < 16; ++s16) hb += sb[s16] * w2[(128 + s16) * 64 + j];
      for (int p = 0; p < 16; ++p)       hb += pb[p]  * w2[(144 + p)   * 64 + j];
      float tu = ew[(64 + j) * 2 + c];
      if (isA) tu = ew[j * 2 + c] - tu;
      s += hb * tu;
    }
    if (isA) s += eb[c];   // fold edge-MLP bias into a' (added once per node)
    Cb[g] = s;
  }
}

// ---------------------------------------------------------------------------
// K1: per-node collapsed GEMM via V_WMMA_F32_16X16X4_F32.
//   One wave = one 16-node tile; K=256 in 64 chunks of 4.
//   A 16x4 f32 layout: lanes 0-15 -> (K=k0,k0+1), lanes 16-31 -> (K=k0+2,k0+3).
//   B 4x16: lane = N column, same K half-split (mirrors C/D layout rule).
//   B staged in LDS as float2 K-pairs; A software-pipelined one chunk ahead.
// ---------------------------------------------------------------------------
__global__ __launch_bounds__(256)
void spell_node_wmma(const float* __restrict__ x,
                     const float* __restrict__ Wpad, const float* __restrict__ Sc,
                     const float* __restrict__ Pc,   const float* __restrict__ Cb,
                     float* __restrict__ Abuf, float* __restrict__ Bbuf,
                     unsigned int* __restrict__ maxEnc)
{
  // Wl[p*16 + col] = (Wpad[2p][col], Wpad[2p+1][col]) : 128 K-pairs x 16 cols.
  __shared__ float2 Wl[128 * 16];            // 16 KB
  __shared__ float  stage[8][256];           // per-wave 16x16 C tile, 8 KB

  const int lane  = threadIdx.x & 31;
  const int wave  = threadIdx.x >> 5;
  const int tile  = blockIdx.x * 8 + wave;
  const bool active = tile < TILES;          // uniform per wave -> EXEC all-1s inside

  // Cooperative staging of the collapsed weights into LDS (once per block).
  for (int i = threadIdx.x; i < 128 * 16; i += 256) {
    const int p = i >> 4, c = i & 15;
    Wl[i] = make_float2(Wpad[(2 * p) * 16 + c], Wpad[(2 * p + 1) * 16 + c]);
  }
  __syncthreads();

  const int mrow  = lane & 15;               // A row / B column / C column
  const int khalf = lane >> 4;               // K sub-pair selector
  const int koff  = khalf * 2;

  v8f acc = {};
  if (active) {
    const int node = tile * 16 + mrow;
    const float* xrow = x + (size_t)node * XDIM;
    // Software pipeline: a_cur holds chunk k0, prefetch k0+4 each iteration.
    v2f a_cur; a_cur.x = xrow[koff]; a_cur.y = xrow[koff + 1];
#pragma unroll 8
    for (int k0 = 0; k0 < 256; k0 += 4) {
      // Prefetch next chunk's A. Last iteration reads cols 256..259 (<901): safe.
      v2f a_nxt; a_nxt.x = xrow[k0 + 4 + koff]; a_nxt.y = xrow[k0 + 5 + koff];
      const float2 bb = Wl[((k0 >> 1) + khalf) * 16 + mrow];  // one ds_load_b64
      v2f b; b.x = bb.x; b.y = bb.y;
      acc = __builtin_amdgcn_wmma_f32_16x16x4_f32(
          /*neg_a=*/false, a_cur, /*neg_b=*/false, b,
          /*c_mod=*/(short)0, acc, /*reuse_a=*/false, /*reuse_b=*/false);
      a_cur = a_nxt;
    }
    // C/D layout: lanes 0-15 rows 0-7, lanes 16-31 rows 8-15; N = lane&15.
#pragma unroll
    for (int r = 0; r < 8; ++r)
      stage[wave][(khalf * 8 + r) * 16 + mrow] = acc[r];
  }
  __syncthreads();
  if (active) {
    // 16 nodes x 8 columns = 128 epilogue entries, 4 per lane.
    for (int e = lane; e < 128; e += 32) {
      const int nl = e >> 3, g = e & 7;
      const int node = tile * 16 + nl;
      const float* xr = x + (size_t)node * XDIM;
      float v = stage[wave][nl * 16 + g];
      int spk = (int)(xr[900] - 1.0f);       // x[:,-1] in {1..4} -> {0..3}
      spk = spk > 2 ? 2 : spk;               // clamp to MAX_SPK-1
      v += Sc[spk * 8 + g] + Cb[g]
         + xr[896] * Pc[0 * 8 + g] + xr[897] * Pc[1 * 8 + g]
         + xr[898] * Pc[2 * 8 + g] + xr[899] * Pc[3 * 8 + g];
      const int idx = (g >> 2) * 2 + (g & 1); // 0,1=face ; 2,3=body
      if ((g & 2) == 0) {
        Abuf[(size_t)node * 4 + idx] = v;                 // a' (bias folded)
      } else {
        Bbuf[(size_t)node * 4 + idx] = v;                 // b (to be maxed)
        maxEnc[(size_t)node * 4 + idx] = 0u;              // init segment-max
      }
    }
  }
}

// ---------------------------------------------------------------------------
// K2: per-edge segment-max via monotone-encoded u32 atomicMax (deterministic).
// ---------------------------------------------------------------------------
__global__ __launch_bounds__(256)
void spell_edge(const int* __restrict__ ei, const int* __restrict__ ea,
                const float* __restrict__ Bbuf, unsigned int* __restrict__ maxEnc)
{
  const int e = blockIdx.x * blockDim.x + threadIdx.x;
  if (e >= N_EDGES) return;
  const int attr = ea[e];
  if (!((attr == 111) || (attr == 0))) return;   // 'valid' mask
  const int src = ei[e];
  const int dst = ei[N_EDGES + e];
  const float4 b = *reinterpret_cast<const float4*>(Bbuf + (size_t)src * 4);
  unsigned int* m = maxEnc + (size_t)dst * 4;
  atomicMax(&m[0], encf(b.x));
  atomicMax(&m[1], encf(b.y));
  atomicMax(&m[2], encf(b.z));
  atomicMax(&m[3], encf(b.w));
}

// ---------------------------------------------------------------------------
// K3: out[i] = sum over blocks of (empty ? 0 : a' + max_b)   (matches isfinite->0)
// ---------------------------------------------------------------------------
__global__ __launch_bounds__(256)
void spell_out(const float* __restrict__ Abuf,
               const unsigned int* __restrict__ maxEnc,
               float* __restrict__ out)
{
  const int i = blockIdx.x * blockDim.x + threadIdx.x;
  if (i >= N_NODES) return;
  const float4 a = *reinterpret_cast<const float4*>(Abuf + (size_t)i * 4);
  const uint4  m = *reinterpret_cast<const uint4*>(maxEnc + (size_t)i * 4);
  const float oF0 = (m.x != 0u) ? (a.x + decf(m.x)) : 0.0f;
  const float oF1 = (m.y != 0u) ? (a.y + decf(m.y)) : 0.0f;
  const float oB0 = (m.z != 0u) ? (a.z + decf(m.z)) : 0.0f;
  const float oB1 = (m.w != 0u) ? (a.w + decf(m.w)) : 0.0f;
  out[(size_t)i * 2 + 0] = oF0 + oB0;
  out[(size_t)i * 2 + 1] = oF1 + oB1;
}

extern "C" void kernel_launch(void* const* d_in, const int* in_sizes, int n_in,
                              void* d_out, int out_size, void* d_ws, size_t ws_size,
                              hipStream_t stream) {
  (void)in_sizes; (void)n_in; (void)out_size; (void)ws_size;
  const float* x   = (const float*)d_in[0];
  const int*   ei  = (const int*)  d_in[1];
  const int*   ea  = (const int*)  d_in[2];
  const float* sw  = (const float*)d_in[3];
  const float* sb  = (const float*)d_in[4];
  const float* pw  = (const float*)d_in[5];
  const float* pb  = (const float*)d_in[6];
  const float* f1w = (const float*)d_in[7];
  const float* f1b = (const float*)d_in[8];
  const float* f2w = (const float*)d_in[9];
  const float* f2b = (const float*)d_in[10];
  const float* few = (const float*)d_in[11];
  const float* feb = (const float*)d_in[12];
  const float* g1w = (const float*)d_in[13];
  const float* g1b = (const float*)d_in[14];
  const float* g2w = (const float*)d_in[15];
  const float* g2b = (const float*)d_in[16];
  const float* gew = (const float*)d_in[17];
  const float* geb = (const float*)d_in[18];

  // Workspace layout (256B-aligned offsets; ~2.42 MB total)
  char* ws = (char*)d_ws;
  float*        Wpad   = (float*)(ws + 0);          // 256*16*4  = 16384
  float*        Sc     = (float*)(ws + 16384);      // 3*8*4     (256B slot)
  float*        Pc     = (float*)(ws + 16640);      // 4*8*4     (256B slot)
  float*        Cb     = (float*)(ws + 16896);      // 8*4       (256B slot)
  float*        Abuf   = (float*)(ws + 17408);      // N*4*4 = 800000
  float*        Bbuf   = (float*)(ws + 17408 + 800000);
  unsigned int* maxEnc = (unsigned int*)(ws + 17408 + 1600000);

  spell_collapse<<<1, 256, 0, stream>>>(sw, sb, pw, pb, f1w, f1b, f2w, f2b,
                                        few, feb, g1w, g1b, g2w, g2b, gew, geb,
                                        Wpad, Sc, Pc, Cb);
  spell_node_wmma<<<(TILES + 7) / 8, 256, 0, stream>>>(x, Wpad, Sc, Pc, Cb,
                                                       Abuf, Bbuf, maxEnc);
  spell_edge<<<(N_EDGES + 255) / 256, 256, 0, stream>>>(ei, ea, Bbuf, maxEnc);
  spell_out<<<(N_NODES + 255) / 256, 256, 0, stream>>>(Abuf, maxEnc, (float*)d_out);
}